// CrossAttention_29472065585752
// MI455X (gfx1250) — compile-verified
//
#include <hip/hip_runtime.h>
#include <hip/hip_bf16.h>
#include <cstdint>

// ---------------------------------------------------------------------------
// CrossAttention on MI455X (gfx1250), f32 end-to-end via V_WMMA_F32_16X16X4_F32,
// with CDNA5 async-to-LDS double-buffered staging for the attention passes.
// Pipeline:
//   1-3) conv1x1 (WMMA GEMM, K=256): K = x1*W1+b1, Q = x1*W2+b2, V = x2*W3+b3
//   4)   stats pass: online (max, sumexp) of kq = K*Q^T per wave (no kq stored)
//   5)   reduce (m,s) pairs -> global (m_g, s_g) per batch-head (global softmax)
//   6)   kqv pass: recompute kq tiles, P = exp(kq - m_g), out = (P*V)/s_g
//   7)   conv1x1: out = kqv*W4 + b4
// ---------------------------------------------------------------------------

typedef float v2f __attribute__((ext_vector_type(2)));
typedef float v8f __attribute__((ext_vector_type(8)));

#define NSP   4096      // H*W
#define CCH   256       // channels
#define DHD   64        // C/HEADS
#define NG    8         // B*HEADS
#define GSTR  (NSP * DHD)   // 262144 elements per head chunk

static __device__ __forceinline__ v8f wmma4(v2f a, v2f b, v8f c) {
  // D(16x16,f32) = A(16x4,f32) x B(4x16,f32) + C
  return __builtin_amdgcn_wmma_f32_16x16x4_f32(
      /*neg_a=*/false, a, /*neg_b=*/false, b,
      /*c_mod=*/(short)0, c, /*reuse_a=*/false, /*reuse_b=*/false);
}

// Async-copy 8KB (contiguous) global -> LDS; 128 threads x 4 b128 each.
// Tracked on ASYNCcnt (4 instructions per wave).
static __device__ __forceinline__ void async_stage_8k(uint32_t lds_base,
                                                      const float* __restrict__ src,
                                                      int tid) {
  uint64_t ga = (uint64_t)(uintptr_t)src + (uint32_t)(tid * 16);
  uint32_t la = lds_base + (uint32_t)(tid * 16);
#pragma unroll
  for (int i = 0; i < 4; ++i) {
    asm volatile("global_load_async_to_lds_b128 %0, %1, off"
                 :: "v"(la), "v"(ga) : "memory");
    ga += 2048;
    la += 2048;
  }
}

#define WAIT_ASYNC(n) asm volatile("s_wait_asynccnt %0" :: "i"(n) : "memory")

static __device__ __forceinline__ uint32_t lds_off32(const void* p) {
  // Generic LDS pointers carry the LDS byte offset in addr[31:0]
  // (shared aperture occupies addr[63:32] only).
  return (uint32_t)(uintptr_t)p;
}

// ---------------------------------------------------------------------------
// conv1x1: Out[b,o,s] = sum_c W[o,c]*X[b,c,s] + bias[o]
// block = 128 threads (4 waves); wave w -> 16 o-rows, block -> 64x64 tile.
// grid = (NSP/64, CCH/64, B)
// ---------------------------------------------------------------------------
__global__ void conv1x1_wmma(const float* __restrict__ X,
                             const float* __restrict__ W,
                             const float* __restrict__ bias,
                             float* __restrict__ Out) {
  const int b    = blockIdx.z;
  const int wave = threadIdx.x >> 5;
  const int lane = threadIdx.x & 31;
  const int r    = lane & 15;         // row/col within 16
  const int h    = lane >> 4;         // half-wave select
  const int obase = blockIdx.y * 64 + wave * 16;
  const int sbase = blockIdx.x * 64;
  const float* Xb = X + (size_t)b * CCH * NSP;
  float*       Ob = Out + (size_t)b * CCH * NSP;

  v8f acc[4] = {};
  for (int kk = 0; kk < CCH / 4; ++kk) {
    const int k0 = kk * 4;
    // A = W tile (16 o-rows x 4 c): lane holds W[obase+r][k0+2h .. +1]
    v2f a = *(const v2f*)(W + (size_t)(obase + r) * CCH + k0 + 2 * h);
#pragma unroll
    for (int t = 0; t < 4; ++t) {
      // B = X tile (4 c-rows x 16 s): lane holds X[k0+2h][scol], X[k0+2h+1][scol]
      const float* xp = Xb + (size_t)(k0 + 2 * h) * NSP + sbase + t * 16 + r;
      v2f bb; bb.x = xp[0]; bb.y = xp[NSP];
      acc[t] = wmma4(a, bb, acc[t]);
    }
  }
#pragma unroll
  for (int t = 0; t < 4; ++t) {
#pragma unroll
    for (int v = 0; v < 8; ++v) {
      const int o = obase + v + 8 * h;                 // D row
      Ob[(size_t)o * NSP + sbase + t * 16 + r] = acc[t][v] + bias[o];
    }
  }
}

// ---------------------------------------------------------------------------
// stats pass: per wave (one 16-row strip of kq for one g), online (max, sumexp)
// over all 4096 columns. kq[i][j] = sum_k K[i][k]*Q[j][k].
// Q j-rows are staged into LDS (8KB stages = 2 j-tiles) via async-to-LDS,
// double-buffered, shared by the 4 waves of the block.
// grid = (256/4, NG), block = 128 (4 waves).
// ---------------------------------------------------------------------------
__global__ void attn_stats(const float* __restrict__ K,
                           const float* __restrict__ Q,
                           float* __restrict__ stats) {
  __shared__ float qs[2][2048];                       // 2 x 8KB stage buffers
  const int g    = blockIdx.y;
  const int tid  = threadIdx.x;
  const int wave = tid >> 5;
  const int lane = tid & 31;
  const int r = lane & 15, h = lane >> 4;
  const int itile = blockIdx.x * 4 + wave;
  const int i0 = itile * 16;
  const float* Kg = K + (size_t)g * GSTR;
  const float* Qg = Q + (size_t)g * GSTR;
  const uint32_t qsb0 = lds_off32(&qs[0][0]);
  const uint32_t qsb1 = lds_off32(&qs[1][0]);

  // Preload A = K-tile (16 rows x 64 k) in A-layout registers (per wave).
  v2f a[16];
#pragma unroll
  for (int kk = 0; kk < 16; ++kk)
    a[kk] = *(const v2f*)(Kg + (size_t)(i0 + r) * DHD + kk * 4 + 2 * h);

  float m = -3.0e38f, s = 0.0f;

  async_stage_8k(qsb0, Qg, tid);                      // stage 0 -> buf0
  for (int st = 0; st < NSP / 32; ++st) {             // 128 stages of 32 rows
    const int cur = st & 1;
    if (st + 1 < NSP / 32) {
      __syncthreads();                                // buf[1-cur] free to overwrite
      async_stage_8k(cur ? qsb0 : qsb1, Qg + (size_t)(st + 1) * 2048, tid);
      WAIT_ASYNC(4);                                  // stage st resident
    } else {
      WAIT_ASYNC(0);
    }
    __syncthreads();                                  // publish stage to all waves

    const float* qt = &qs[cur][0];
#pragma unroll
    for (int jt = 0; jt < 2; ++jt) {
      v8f acc = {};
#pragma unroll
      for (int kk = 0; kk < 16; ++kk) {
        // B[k][j] = Q[jrow][k], read from LDS (ds_load_b64)
        v2f bb = *(const v2f*)(qt + (jt * 16 + r) * DHD + kk * 4 + 2 * h);
        acc = wmma4(a[kk], bb, acc);
      }
      float tm = acc[0];
#pragma unroll
      for (int v = 1; v < 8; ++v) tm = fmaxf(tm, acc[v]);
      const float nm = fmaxf(m, tm);
      float ls = s * __expf(m - nm);
#pragma unroll
      for (int v = 0; v < 8; ++v) ls += __expf(acc[v] - nm);
      m = nm; s = ls;
    }
  }
  // wave reduction of (m, s)
#pragma unroll
  for (int off = 16; off >= 1; off >>= 1) {
    const float om = __shfl_xor(m, off, 32);
    const float os = __shfl_xor(s, off, 32);
    const float nm = fmaxf(m, om);
    s = s * __expf(m - nm) + os * __expf(om - nm);
    m = nm;
  }
  if (lane == 0) {
    stats[((size_t)g * 256 + itile) * 2 + 0] = m;
    stats[((size_t)g * 256 + itile) * 2 + 1] = s;
  }
}

// one wave per g: combine 256 (m,s) pairs -> global (m_g, s_g)
__global__ void attn_reduce(const float* __restrict__ stats,
                            float* __restrict__ gms) {
  const int g = blockIdx.x;
  const int lane = threadIdx.x;
  float m = -3.0e38f, s = 0.0f;
  for (int t = lane; t < 256; t += 32) {
    const float tm = stats[((size_t)g * 256 + t) * 2 + 0];
    const float ts = stats[((size_t)g * 256 + t) * 2 + 1];
    const float nm = fmaxf(m, tm);
    s = s * __expf(m - nm) + ts * __expf(tm - nm);
    m = nm;
  }
#pragma unroll
  for (int off = 16; off >= 1; off >>= 1) {
    const float om = __shfl_xor(m, off, 32);
    const float os = __shfl_xor(s, off, 32);
    const float nm = fmaxf(m, om);
    s = s * __expf(m - nm) + os * __expf(om - nm);
    m = nm;
  }
  if (lane == 0) { gms[g * 2 + 0] = m; gms[g * 2 + 1] = s; }
}

// ---------------------------------------------------------------------------
// kqv pass: out[i][:] = (1/s_g) * sum_j exp(kq[i][j]-m_g) * V[j][:]
// Computes kq^T tiles (A=Q rows, B=K rows) so the D-layout result converts to
// the A-operand of P*V with a half-wave shfl_xor(16) + per-lane selects.
// Q and V j-rows staged into LDS via async-to-LDS, double buffered.
// grid = (256/4, NG), block = 128 (4 waves).
// ---------------------------------------------------------------------------
__global__ void attn_kqv(const float* __restrict__ K,
                         const float* __restrict__ Q,
                         const float* __restrict__ V,
                         const float* __restrict__ gms,
                         float* __restrict__ O) {
  __shared__ float qs[2][2048];                       // 2 x 8KB (Q rows)
  __shared__ float vs[2][2048];                       // 2 x 8KB (V rows)
  const int g    = blockIdx.y;
  const int tid  = threadIdx.x;
  const int wave = tid >> 5;
  const int lane = tid & 31;
  const int r = lane & 15, h = lane >> 4;
  const int itile = blockIdx.x * 4 + wave;
  const int i0 = itile * 16;
  const float* Kg = K + (size_t)g * GSTR;
  const float* Qg = Q + (size_t)g * GSTR;
  const float* Vg = V + (size_t)g * GSTR;
  const float gm = gms[g * 2 + 0];
  const float gs = gms[g * 2 + 1];
  const uint32_t qsb[2] = { lds_off32(&qs[0][0]), lds_off32(&qs[1][0]) };
  const uint32_t vsb[2] = { lds_off32(&vs[0][0]), lds_off32(&vs[1][0]) };

  // Preload B' = K-tile in B-layout: B'[k][col i] = K[i0+i][k]
  v2f bK[16];
#pragma unroll
  for (int kk = 0; kk < 16; ++kk)
    bK[kk] = *(const v2f*)(Kg + (size_t)(i0 + r) * DHD + kk * 4 + 2 * h);

  v8f accv[4] = {};

  async_stage_8k(qsb[0], Qg, tid);
  async_stage_8k(vsb[0], Vg, tid);
  for (int st = 0; st < NSP / 32; ++st) {             // 128 stages of 32 rows
    const int cur = st & 1;
    if (st + 1 < NSP / 32) {
      __syncthreads();                                // bufs[1-cur] free
      async_stage_8k(qsb[cur ^ 1], Qg + (size_t)(st + 1) * 2048, tid);
      async_stage_8k(vsb[cur ^ 1], Vg + (size_t)(st + 1) * 2048, tid);
      WAIT_ASYNC(8);                                  // stage st resident
    } else {
      WAIT_ASYNC(0);
    }
    __syncthreads();

    const float* qt = &qs[cur][0];
    const float* vt = &vs[cur][0];
#pragma unroll
    for (int jt = 0; jt < 2; ++jt) {
      // D = kq^T tile: D[jl][il] = sum_k Q[jrow][k] * K[i0+il][k]
      v8f d = {};
#pragma unroll
      for (int kk = 0; kk < 16; ++kk) {
        v2f aQ = *(const v2f*)(qt + (jt * 16 + r) * DHD + kk * 4 + 2 * h);
        d = wmma4(aQ, bK[kk], d);
      }
      // P[i=r][j = v + 8h] = exp(d[v] - m_g)   (this lane's view)
      float pv[8], sw[8];
#pragma unroll
      for (int v = 0; v < 8; ++v) pv[v] = __expf(d[v] - gm);
#pragma unroll
      for (int v = 0; v < 8; ++v) sw[v] = __shfl_xor(pv[v], 16, 32);
      // P[r][j] = ((j>>3)==h) ? pv[j&7] : sw[j&7]

      // Second GEMM: accv[nt] += A2(P tile) x B2(V rows of this j-tile)
#pragma unroll
      for (int t = 0; t < 4; ++t) {
        const int jx0 = 4 * t,     jx1 = 4 * t + 2;   // a2.x for h=0 / h=1
        const int jy0 = 4 * t + 1, jy1 = 4 * t + 3;   // a2.y for h=0 / h=1
        const float ax0 = ((jx0 >> 3) == 0) ? pv[jx0 & 7] : sw[jx0 & 7];
        const float ax1 = ((jx1 >> 3) == 1) ? pv[jx1 & 7] : sw[jx1 & 7];
        const float ay0 = ((jy0 >> 3) == 0) ? pv[jy0 & 7] : sw[jy0 & 7];
        const float ay1 = ((jy1 >> 3) == 1) ? pv[jy1 & 7] : sw[jy1 & 7];
        v2f a2; a2.x = h ? ax1 : ax0; a2.y = h ? ay1 : ay0;
        const float* vp = vt + (jt * 16 + 4 * t + 2 * h) * DHD;
#pragma unroll
        for (int nt = 0; nt < 4; ++nt) {
          v2f bv; bv.x = vp[nt * 16 + r]; bv.y = vp[DHD + nt * 16 + r];
          accv[nt] = wmma4(a2, bv, accv[nt]);
        }
      }
    }
  }

  const float inv = 1.0f / gs;
#pragma unroll
  for (int nt = 0; nt < 4; ++nt) {
#pragma unroll
    for (int v = 0; v < 8; ++v) {
      const int row = i0 + v + 8 * h;
      O[(size_t)g * GSTR + (size_t)row * DHD + nt * 16 + r] = accv[nt][v] * inv;
    }
  }
}

// ---------------------------------------------------------------------------
extern "C" void kernel_launch(void* const* d_in, const int* in_sizes, int n_in,
                              void* d_out, int out_size, void* d_ws, size_t ws_size,
                              hipStream_t stream) {
  const float* x1 = (const float*)d_in[0];
  const float* x2 = (const float*)d_in[1];
  const float* W1 = (const float*)d_in[2];
  const float* b1 = (const float*)d_in[3];
  const float* W2 = (const float*)d_in[4];
  const float* b2 = (const float*)d_in[5];
  const float* W3 = (const float*)d_in[6];
  const float* b3 = (const float*)d_in[7];
  const float* W4 = (const float*)d_in[8];
  const float* b4 = (const float*)d_in[9];
  float* out = (float*)d_out;

  float* ws = (float*)d_ws;
  const size_t PER = 2ull * CCH * NSP;   // elements per [B,C,N] tensor
  float* Kb = ws;
  float* Qb = ws + PER;
  float* Vb = ws + 2 * PER;
  float* KQ = ws + 3 * PER;
  float* st = ws + 4 * PER;              // 2048 (m,s) pairs
  float* gm = st + 4096;                 // 8 (m,s) pairs

  const dim3 cgrid(NSP / 64, CCH / 64, 2);
  const dim3 cblk(128);
  conv1x1_wmma<<<cgrid, cblk, 0, stream>>>(x1, W1, b1, Kb);
  conv1x1_wmma<<<cgrid, cblk, 0, stream>>>(x1, W2, b2, Qb);
  conv1x1_wmma<<<cgrid, cblk, 0, stream>>>(x2, W3, b3, Vb);
  attn_stats <<<dim3(64, NG), 128, 0, stream>>>(Kb, Qb, st);
  attn_reduce<<<NG, 32, 0, stream>>>(st, gm);
  attn_kqv   <<<dim3(64, NG), 128, 0, stream>>>(Kb, Qb, Vb, gm, KQ);
  conv1x1_wmma<<<cgrid, cblk, 0, stream>>>(KQ, W4, b4, out);
}